// RHD_1649267442080
// MI455X (gfx1250) — compile-verified
//
#include <hip/hip_runtime.h>
#include <math.h>

typedef __attribute__((ext_vector_type(2))) float v2f;
typedef __attribute__((ext_vector_type(8))) float v8f;
typedef __attribute__((ext_vector_type(4))) unsigned int v4u;
typedef __attribute__((ext_vector_type(4))) int v4i;
typedef __attribute__((ext_vector_type(8))) int v8i;

#define Hh 512
#define Ww 512
#define PLANE (512 * 512)      // 262144 = 2^18
#define NPIX  (8 * 512 * 512)  // 2097152
#define INF_F 1000000000000.0f

// ---------------------------------------------------------------------------
// TDM helpers: 2D tile <-> LDS DMA (tensor_load_to_lds / tensor_store_from_lds)
// Descriptor per CDNA5 ISA ch.8: group0 = {count, lds_addr, global_addr, type},
// group1 = {mask/data_size, tensor dims, tile dims, dim0 stride}, groups 2/3 = 0.
// ---------------------------------------------------------------------------
__device__ __forceinline__ void tdm_desc_512x32(const float* gtile, unsigned ldsa,
                                                v4u& g0, v8i& g1) {
  unsigned long long ga = (unsigned long long)gtile;
  g0.x = 1u;                                   // count=1 (valid user descriptor)
  g0.y = ldsa;                                 // lds_addr (bytes)
  g0.z = (unsigned)(ga & 0xFFFFFFFFu);         // global_addr[31:0]
  g0.w = (unsigned)((ga >> 32) & 0x01FFFFFFu)  // global_addr[56:32]
       | (2u << 30);                           // type = 2 ("image")
  g1[0] = 0x00020000;                          // workgroup_mask=0, data_size=2 (4B)
  g1[1] = (512 << 16);                         // tensor_dim0[15:0]=512 (at bits 63:48)
  g1[2] = (512 << 16);                         // tensor_dim0 hi=0; tensor_dim1[15:0]=512
  g1[3] = (32 << 16);                          // tensor_dim1 hi=0; tile_dim0=32
  g1[4] = 512;                                 // tile_dim1=512; tile_dim2=0
  g1[5] = 512;                                 // tensor_dim0_stride[31:0]=512 elements
  g1[6] = 0;                                   // stride hi=0; tensor_dim1_stride lo=0
  g1[7] = 0;
}

__device__ __forceinline__ void tdm_load_512x32(const float* gtile, unsigned ldsa) {
  v4u g0; v8i g1;
  tdm_desc_512x32(gtile, ldsa, g0, g1);
  v4i gz = {0, 0, 0, 0};
#if __has_include(<hip/amd_detail/amd_gfx1250_TDM.h>)
  v8i z8 = {0, 0, 0, 0, 0, 0, 0, 0};
  __builtin_amdgcn_tensor_load_to_lds(g0, g1, gz, gz, z8, 0);
#else
  __builtin_amdgcn_tensor_load_to_lds(g0, g1, gz, gz, 0);
#endif
  __builtin_amdgcn_s_wait_tensorcnt(0);
}

__device__ __forceinline__ void tdm_store_512x32(float* gtile, unsigned ldsa) {
  v4u g0; v8i g1;
  tdm_desc_512x32(gtile, ldsa, g0, g1);
  v4i gz = {0, 0, 0, 0};
#if __has_include(<hip/amd_detail/amd_gfx1250_TDM.h>)
  v8i z8 = {0, 0, 0, 0, 0, 0, 0, 0};
  __builtin_amdgcn_tensor_store_from_lds(g0, g1, gz, gz, z8, 0);
#else
  __builtin_amdgcn_tensor_store_from_lds(g0, g1, gz, gz, 0);
#endif
  __builtin_amdgcn_s_wait_tensorcnt(0);
}

// ---------------------------------------------------------------------------
// Pass 1: 1D squared EDT along W for both masks, O(n) via wave scans.
// One wave (32 lanes) per row; 8192 rows total (2 masks x 8 imgs x 512 rows).
// f in {0, INF} => result = (distance to nearest background)^2, exact vs ref.
// ---------------------------------------------------------------------------
__global__ __launch_bounds__(256) void edt_pass1(const float* __restrict__ outp,
                                                 const int* __restrict__ lab,
                                                 float* __restrict__ gtbuf,
                                                 float* __restrict__ segbuf) {
  const int tid  = blockIdx.x * blockDim.x + threadIdx.x;
  const int lane = threadIdx.x & 31;
  const int wrow = tid >> 5;           // 0..8191
  const int m    = wrow >> 12;         // 0 = gt mask, 1 = seg mask
  const int r    = wrow & 4095;
  const int b    = r >> 9;
  const int h    = r & 511;
  const size_t base = ((size_t)(b * 2 + 1) * Hh + h) * Ww;  // channel-1 row

  // background flags, strided ownership x = k*32 + lane (coalesced loads)
  unsigned bgmask = 0u;
  if (m == 0) {
#pragma unroll
    for (int k = 0; k < 16; ++k) {
      int x = (k << 5) + lane;
      if (lab[base + x] <= 0) bgmask |= (1u << k);
    }
  } else {
#pragma unroll
    for (int k = 0; k < 16; ++k) {
      int x = (k << 5) + lane;
      if (!(outp[base + x] > 0.5f)) bgmask |= (1u << k);
    }
  }

  // forward prefix-max of (bg ? x : -1)  -> nearest background to the left
  int fwd[16];
  int carry = -1;
#pragma unroll
  for (int k = 0; k < 16; ++k) {
    int x = (k << 5) + lane;
    int v = ((bgmask >> k) & 1) ? x : -1;
#pragma unroll
    for (int off = 1; off < 32; off <<= 1) {
      int t = __shfl_up(v, off, 32);
      if (lane >= off) v = (v > t) ? v : t;
    }
    v = (v > carry) ? v : carry;
    carry = __shfl(v, 31, 32);
    fwd[k] = v;
  }

  // backward suffix-min of (bg ? x : BIG) -> nearest background to the right
  const int BIG = 1 << 20;
  float* outbuf = (m == 0) ? gtbuf : segbuf;
  const size_t obase = (size_t)b * PLANE + (size_t)h * Ww;
  int carryB = BIG;
#pragma unroll
  for (int k = 15; k >= 0; --k) {
    int x = (k << 5) + lane;
    int v = ((bgmask >> k) & 1) ? x : BIG;
#pragma unroll
    for (int off = 1; off < 32; off <<= 1) {
      int t = __shfl_down(v, off, 32);
      if (lane + off < 32) v = (v < t) ? v : t;
    }
    v = (v < carryB) ? v : carryB;
    carryB = __shfl(v, 0, 32);
    int a  = fwd[k];
    int d1 = (a >= 0) ? (x - a) : BIG;
    int d2 = (v < BIG) ? (v - x) : BIG;
    int d  = (d1 < d2) ? d1 : d2;
    outbuf[obase + x] = (d >= BIG) ? INF_F : (float)(d * d);
  }
}

// ---------------------------------------------------------------------------
// Pass 2: brute-force min-plus along H, in-place.  Block = 256 thr (8 waves).
// The 512x32 column tile (64 KB) is staged global<->LDS by the Tensor Data
// Mover (one DMA per direction per block), tracked with TENSORcnt.  One wave
// per column, lane owns 16 outputs (h = lane + 32*m); j unrolled x2 so the
// reduction can fuse to v_min3.
// ---------------------------------------------------------------------------
__global__ __launch_bounds__(256) void edt_pass2(float* __restrict__ gtbuf,
                                                 float* __restrict__ segbuf) {
  __shared__ float tile[512 * 32];     // 64 KB
  const int bi = blockIdx.x;           // 0..255
  const int m  = bi >> 7;
  const int b  = (bi >> 4) & 7;
  const int t0 = (bi & 15) << 5;       // column tile start
  float* buf = (m == 0) ? gtbuf : segbuf;
  float* gtile = buf + (size_t)b * PLANE + t0;
  const unsigned ldsa = (unsigned)(unsigned long long)&tile[0];
  const int tid = threadIdx.x;

  if (tid == 0) tdm_load_512x32(gtile, ldsa);   // TDM: global -> LDS, wait tensorcnt
  __syncthreads();

  const int wv = tid >> 5;
  const int lane = tid & 31;
#pragma unroll 1
  for (int cc = 0; cc < 4; ++cc) {
    const int col = (wv << 2) + cc;
    float minv[16], qf[16];
#pragma unroll
    for (int mm = 0; mm < 16; ++mm) {
      minv[mm] = 3.0e38f;
      qf[mm]   = (float)(lane + (mm << 5));   // q = h - j, starts at h
    }
    for (int j = 0; j < 512; j += 2) {
      float ga = tile[j * 32 + col];          // LDS broadcast, no conflicts
      float gb = tile[(j + 1) * 32 + col];
#pragma unroll
      for (int mm = 0; mm < 16; ++mm) {
        float q0 = qf[mm];
        float q1 = q0 - 1.0f;
        float c0 = fmaf(q0, q0, ga);
        float c1 = fmaf(q1, q1, gb);
        minv[mm] = fminf(minv[mm], fminf(c0, c1));   // -> v_min3 candidate
        qf[mm] = q0 - 2.0f;
      }
    }
    // only this wave reads/writes this column: safe in-place overwrite
#pragma unroll
    for (int mm = 0; mm < 16; ++mm)
      tile[(lane + (mm << 5)) * 32 + col] = minv[mm];
  }
  __syncthreads();

  if (tid == 0) tdm_store_512x32(gtile, ldsa);  // TDM: LDS -> global, wait tensorcnt
}

// ---------------------------------------------------------------------------
// Elementwise loss terms + deterministic per-block partial sums.
// 512 blocks x 256 threads x 16 pixels.
// ---------------------------------------------------------------------------
__global__ __launch_bounds__(256) void loss_partials(const float* __restrict__ outp,
                                                     const int* __restrict__ lab,
                                                     const float* __restrict__ gtbuf,
                                                     const float* __restrict__ segbuf,
                                                     float* __restrict__ part) {
  __shared__ float red[8][5];
  const int tid = threadIdx.x;
  float hd = 0.f, inter = 0.f, ssq = 0.f, lsq = 0.f, ce = 0.f;
  const int base = blockIdx.x * 4096;
#pragma unroll 4
  for (int i = 0; i < 16; ++i) {
    int p   = base + tid + (i << 8);
    int b   = p >> 18;
    int off = p & (PLANE - 1);
    size_t o1 = ((size_t)(b * 2 + 1)) * PLANE + off;
    size_t o0 = ((size_t)(b * 2 + 0)) * PLANE + off;
    float s  = outp[o1];
    float p0 = outp[o0];
    int   lv = lab[o1];
    float lf = (float)lv;
    float gt = gtbuf[p];
    float sg = segbuf[p];
    float d  = s - lf;
    hd    += d * d * (sg + gt);
    inter += s * lf;
    ssq   += s * s;
    lsq   += lf;                       // lf*lf == lf for 0/1 labels
    // log_softmax over the 2 probability channels, pick channel lv
    float mx  = fmaxf(p0, s);
    float lse = mx + logf(expf(p0 - mx) + expf(s - mx));
    float pc  = (lv > 0) ? s : p0;
    ce += (lse - pc);
  }
  float vals[5] = {hd, inter, ssq, lsq, ce};
  const int lane = tid & 31, wv = tid >> 5;
#pragma unroll
  for (int q = 0; q < 5; ++q) {
    float v = vals[q];
    for (int off = 16; off > 0; off >>= 1) v += __shfl_down(v, off, 32);
    if (lane == 0) red[wv][q] = v;
  }
  __syncthreads();
  if (tid == 0) {
#pragma unroll
    for (int q = 0; q < 5; ++q) {
      float s = red[0][q];
      for (int w = 1; w < 8; ++w) s += red[w][q];
      part[q * 512 + blockIdx.x] = s;
    }
  }
}

// ---------------------------------------------------------------------------
// Final reduction: one wave.  Each quantity's 512 partials are summed on the
// matrix pipe: D = A(16x4) x ones(4x16) + C, chained over 8 chunks of 64.
// Column N=0 of D lives in lanes 0 (M=0..7) and 16 (M=8..15).
// ---------------------------------------------------------------------------
__global__ __launch_bounds__(32) void loss_final(const float* __restrict__ part,
                                                 float* __restrict__ out) {
  __shared__ float lds[512];
  const int lane = threadIdx.x;
  float tot[5];
  v2f bones; bones.x = 1.0f; bones.y = 1.0f;
  const int ml   = lane & 15;
  const int koff = (lane >> 4) << 1;   // A layout: lanes 16..31 hold K=2,3
#pragma unroll 1
  for (int q = 0; q < 5; ++q) {
#pragma unroll
    for (int i = 0; i < 16; ++i)
      lds[lane + (i << 5)] = part[q * 512 + lane + (i << 5)];
    __syncthreads();
    v8f acc = {};
#pragma unroll
    for (int t = 0; t < 8; ++t) {
      int bidx = t * 64 + ml * 4 + koff;
      v2f a; a.x = lds[bidx]; a.y = lds[bidx + 1];
      // (neg_a, A, neg_b, B, c_mod, C, reuse_a, reuse_b)
      acc = __builtin_amdgcn_wmma_f32_16x16x4_f32(false, a, false, bones,
                                                  (short)0, acc, false, false);
    }
    float s = acc[0] + acc[1] + acc[2] + acc[3] + acc[4] + acc[5] + acc[6] + acc[7];
    tot[q] = __shfl(s, 0, 32) + __shfl(s, 16, 32);
    __syncthreads();
  }
  const float invN = 1.0f / 2097152.0f;   // exact power of two
  float ce_loss = tot[4] * invN;
  float hd_loss = tot[0] * invN;
  float dice = 1.0f - (2.0f * tot[1] + 1e-6f) / (tot[2] + tot[3] + 1e-6f);
  float loss = (ce_loss + dice) + 0.5f * hd_loss;  // LAM=1, 1-ALPHA=0.5
  if (lane == 0) out[0] = loss;
}

// ---------------------------------------------------------------------------
extern "C" void kernel_launch(void* const* d_in, const int* in_sizes, int n_in,
                              void* d_out, int out_size, void* d_ws, size_t ws_size,
                              hipStream_t stream) {
  (void)in_sizes; (void)n_in; (void)out_size; (void)ws_size;
  const float* outp = (const float*)d_in[0];   // softmax probs (8,2,512,512)
  const int*   lab  = (const int*)d_in[1];     // labels        (8,2,512,512)
  float* gtbuf  = (float*)d_ws;                // 8 MB
  float* segbuf = gtbuf + NPIX;                // 8 MB
  float* part   = segbuf + NPIX;               // 512*5 floats
  float* out    = (float*)d_out;

  edt_pass1<<<1024, 256, 0, stream>>>(outp, lab, gtbuf, segbuf);
  edt_pass2<<<256, 256, 0, stream>>>(gtbuf, segbuf);
  loss_partials<<<512, 256, 0, stream>>>(outp, lab, gtbuf, segbuf, part);
  loss_final<<<1, 32, 0, stream>>>(part, out);
}